// DeformConvBlock_68101001445874
// MI455X (gfx1250) — compile-verified
//
#include <hip/hip_runtime.h>
#include <hip/hip_bf16.h>

#define HH 128
#define WW 128
#define CC 64
#define BB 4
#define NPIX (BB*HH*WW)          // 65536
#define KTOT 576                 // 64 * 9
#define TENSOR_ELEMS (NPIX*CC)   // 4,194,304 floats = 16 MiB

typedef float v2f __attribute__((ext_vector_type(2)));
typedef float v8f __attribute__((ext_vector_type(8)));

// ---------------------------------------------------------------------------
// NCHW -> NHWC transpose (one time, L2 resident afterwards)
// ---------------------------------------------------------------------------
__global__ void nchw_to_nhwc_kernel(const float* __restrict__ x, float* __restrict__ xn) {
    int idx = blockIdx.x * 256 + threadIdx.x;        // over NHWC elements
    int c   = idx & 63;
    int pid = idx >> 6;
    int b   = pid >> 14;
    int hw  = pid & 16383;
    xn[idx] = x[((size_t)(b * CC + c) << 14) + hw];
}

// ---------------------------------------------------------------------------
// Pack deform weights into per-lane WMMA B-fragment order.
// K' = kk*64 + cin.  Fragment (g = K-step, t = N-tile):
//   dword index = (((g*4 + t)*32 + lane)*2 + d)
//   value       = W[o = t*16 + (lane&15)][K = g*4 + 2*(lane>>4) + d]
// ---------------------------------------------------------------------------
__global__ void pack_weights_kernel(const float* __restrict__ dw, float* __restrict__ packed) {
    int idx = blockIdx.x * 256 + threadIdx.x;        // 7 * 36864 = 258048 total
    int layer = idx / 36864;
    int r = idx % 36864;
    int d = r & 1;  r >>= 1;
    int l = r & 31; r >>= 5;
    int t = r & 3;  r >>= 2;
    int g = r;                                       // 0..143
    int o  = t * 16 + (l & 15);
    int Kp = g * 4 + 2 * (l >> 4) + d;               // 0..575
    int kk = Kp >> 6;                                // tap index
    int ci = Kp & 63;                                // input channel
    packed[idx] = dw[(size_t)layer * 36864 + (size_t)(o * 64 + ci) * 9 + kk];
}

__global__ void zero_stats_kernel(float* __restrict__ stats) {
    if (threadIdx.x < 128) stats[threadIdx.x] = 0.f;
}

// ---------------------------------------------------------------------------
// Offset conv: 3x3, 64 -> 18 channels, NHWC input, [pid][18] output.
// Weight indices are lane-uniform -> scalar loads; ~1.4 GFLOP/layer on VALU.
// ---------------------------------------------------------------------------
__global__ void offset_conv_kernel(const float* __restrict__ xn,
                                   const float* __restrict__ ow,
                                   const float* __restrict__ ob,
                                   float* __restrict__ off) {
    int pid = blockIdx.x * 256 + threadIdx.x;        // 65536 threads
    int b  = pid >> 14;
    int hw = pid & 16383;
    int h  = hw >> 7;
    int w  = hw & 127;

    float acc[18];
#pragma unroll
    for (int j = 0; j < 18; ++j) acc[j] = ob[j];

    for (int kh = 0; kh < 3; ++kh) {
        int y = h - 1 + kh;
        if (y < 0 || y >= HH) continue;
        for (int kw = 0; kw < 3; ++kw) {
            int x = w - 1 + kw;
            if (x < 0 || x >= WW) continue;
            int kkb = kh * 3 + kw;
            const float4* xp = (const float4*)(xn + (size_t)((b * HH + y) * WW + x) * CC);
#pragma unroll 4
            for (int c4 = 0; c4 < 16; ++c4) {
                float4 xv = xp[c4];
#pragma unroll
                for (int j = 0; j < 18; ++j) {
                    const float* wr = ow + (size_t)(j * 64 + c4 * 4) * 9 + kkb;
                    acc[j] += xv.x * wr[0] + xv.y * wr[9] + xv.z * wr[18] + xv.w * wr[27];
                }
            }
        }
    }
#pragma unroll
    for (int j = 0; j < 18; ++j) off[pid * 18 + j] = acc[j];
}

// ---------------------------------------------------------------------------
// Deformable conv GEMM: per wave, M-tile = 16 pixels, N = 64 (all Cout),
// K = 576 processed as 9 taps x 64 channels.  fp32 WMMA 16x16x4.
// Also accumulates per-channel sum / sumsq for batchnorm.
// ---------------------------------------------------------------------------
__launch_bounds__(256)
__global__ void deform_wmma_kernel(const float* __restrict__ xn,
                                   const float* __restrict__ off,
                                   const float* __restrict__ Bpack,
                                   float* __restrict__ y,
                                   float* __restrict__ stats) {
    __shared__ float Alds[8 * 16 * 68];              // stride 68 -> conflict-free
    __shared__ float sstat[128];

    const int tid  = threadIdx.x;
    const int lane = tid & 31;
    const int wv   = tid >> 5;                       // wave 0..7
    const int half = lane >> 4;
    const int p    = lane & 15;
    const int tile0 = blockIdx.x * 128 + wv * 16;    // first pixel of this wave's tile
    const int pid  = tile0 + p;
    const int b    = pid >> 14;
    const int hw   = pid & 16383;
    const int h    = hw >> 7;
    const int w    = hw & 127;

    if (tid < 128) sstat[tid] = 0.f;
    float* Aw = &Alds[wv * (16 * 68)];
    const v2f* __restrict__ Bp = (const v2f*)Bpack;

    v8f acc0 = {0,0,0,0,0,0,0,0};
    v8f acc1 = {0,0,0,0,0,0,0,0};
    v8f acc2 = {0,0,0,0,0,0,0,0};
    v8f acc3 = {0,0,0,0,0,0,0,0};

    __syncthreads();

    for (int kk = 0; kk < 9; ++kk) {
        // ---- bilinear-sample 16 pixels x 64 channels for this tap into LDS ----
        float dy = off[pid * 18 + 2 * kk];
        float dx = off[pid * 18 + 2 * kk + 1];
        float py = dy + (float)(h - 1 + (kk / 3));
        float px = dx + (float)(w - 1 + (kk % 3));
        float y0f = floorf(py), x0f = floorf(px);
        float ly = py - y0f, lx = px - x0f;
        int y0 = (int)y0f, x0 = (int)x0f;
        int y1 = y0 + 1,  x1 = x0 + 1;
        float ly1 = 1.f - ly, lx1 = 1.f - lx;
        float vy0 = (y0 >= 0 && y0 < HH) ? 1.f : 0.f;
        float vy1 = (y1 >= 0 && y1 < HH) ? 1.f : 0.f;
        float vx0 = (x0 >= 0 && x0 < WW) ? 1.f : 0.f;
        float vx1 = (x1 >= 0 && x1 < WW) ? 1.f : 0.f;
        int y0c = min(max(y0, 0), HH - 1), y1c = min(max(y1, 0), HH - 1);
        int x0c = min(max(x0, 0), WW - 1), x1c = min(max(x1, 0), WW - 1);
        float w00 = ly1 * lx1 * vy0 * vx0;
        float w01 = ly1 * lx  * vy0 * vx1;
        float w10 = ly  * lx1 * vy1 * vx0;
        float w11 = ly  * lx  * vy1 * vx1;
        const float4* c00 = (const float4*)(xn + (size_t)((b*HH + y0c)*WW + x0c)*CC + half*32);
        const float4* c01 = (const float4*)(xn + (size_t)((b*HH + y0c)*WW + x1c)*CC + half*32);
        const float4* c10 = (const float4*)(xn + (size_t)((b*HH + y1c)*WW + x0c)*CC + half*32);
        const float4* c11 = (const float4*)(xn + (size_t)((b*HH + y1c)*WW + x1c)*CC + half*32);
        float4* dstl = (float4*)&Aw[p * 68 + half * 32];
#pragma unroll
        for (int c4 = 0; c4 < 8; ++c4) {
            float4 a0 = c00[c4], a1 = c01[c4], a2 = c10[c4], a3 = c11[c4];
            float4 r;
            r.x = a0.x*w00 + a1.x*w01 + a2.x*w10 + a3.x*w11;
            r.y = a0.y*w00 + a1.y*w01 + a2.y*w10 + a3.y*w11;
            r.z = a0.z*w00 + a1.z*w01 + a2.z*w10 + a3.z*w11;
            r.w = a0.w*w00 + a1.w*w01 + a2.w*w10 + a3.w*w11;
            dstl[c4] = r;
        }
        __syncthreads();

        // ---- 16 K-steps x 4 N-tiles of V_WMMA_F32_16X16X4_F32 ----
#pragma unroll 4
        for (int s = 0; s < 16; ++s) {
            v2f a;
            a.x = Aw[p * 68 + s * 4 + 2 * half];
            a.y = Aw[p * 68 + s * 4 + 2 * half + 1];
            int g = kk * 16 + s;
            v2f b0 = Bp[(g * 4 + 0) * 32 + lane];
            v2f b1 = Bp[(g * 4 + 1) * 32 + lane];
            v2f b2 = Bp[(g * 4 + 2) * 32 + lane];
            v2f b3 = Bp[(g * 4 + 3) * 32 + lane];
            acc0 = __builtin_amdgcn_wmma_f32_16x16x4_f32(false, a, false, b0, (short)0, acc0, false, false);
            acc1 = __builtin_amdgcn_wmma_f32_16x16x4_f32(false, a, false, b1, (short)0, acc1, false, false);
            acc2 = __builtin_amdgcn_wmma_f32_16x16x4_f32(false, a, false, b2, (short)0, acc2, false, false);
            acc3 = __builtin_amdgcn_wmma_f32_16x16x4_f32(false, a, false, b3, (short)0, acc3, false, false);
        }
        __syncthreads();
    }

    // ---- write D (NHWC) + per-channel stats via LDS then one global atomic ----
    v8f accs[4] = {acc0, acc1, acc2, acc3};
#pragma unroll
    for (int t = 0; t < 4; ++t) {
        v8f A = accs[t];
        float s1 = 0.f, s2 = 0.f;
        int ch = t * 16 + p;
#pragma unroll
        for (int r = 0; r < 8; ++r) {
            float v = A[r];
            int prow = tile0 + r + 8 * half;         // C/D layout: M = r + 8*(lane>>4)
            y[(size_t)prow * 64 + ch] = v;
            s1 += v; s2 += v * v;
        }
        atomicAdd(&sstat[ch], s1);
        atomicAdd(&sstat[64 + ch], s2);
    }
    __syncthreads();
    if (tid < 128) atomicAdd(&stats[tid], sstat[tid]);
}

// ---------------------------------------------------------------------------
// BatchNorm (stats over B,H,W) + affine + ReLU, in place on NHWC tensor.
// ---------------------------------------------------------------------------
__global__ void norm_relu_kernel(float* __restrict__ yv, const float* __restrict__ stats,
                                 const float* __restrict__ g, const float* __restrict__ bt) {
    int idx = blockIdx.x * 256 + threadIdx.x;        // 4,194,304 elements
    int c = idx & 63;
    const float inv_n = 1.0f / (float)NPIX;
    float mean = stats[c] * inv_n;
    float var  = stats[64 + c] * inv_n - mean * mean;
    float v = (yv[idx] - mean) * rsqrtf(var + 1e-5f);
    v = g[c] * v + bt[c];
    yv[idx] = fmaxf(v, 0.f);
}

__global__ void add_kernel(const float* __restrict__ a, const float* __restrict__ b,
                           float* __restrict__ o) {
    int idx = blockIdx.x * 256 + threadIdx.x;        // over float4s: 1,048,576
    float4 av = ((const float4*)a)[idx];
    float4 bv = ((const float4*)b)[idx];
    float4 r; r.x = av.x + bv.x; r.y = av.y + bv.y; r.z = av.z + bv.z; r.w = av.w + bv.w;
    ((float4*)o)[idx] = r;
}

// Final 3-way add + NHWC -> NCHW transpose into d_out.
__global__ void combine_out_kernel(const float* __restrict__ a, const float* __restrict__ b,
                                   const float* __restrict__ c, float* __restrict__ out) {
    int idx = blockIdx.x * 256 + threadIdx.x;        // NCHW index
    int bn = idx >> 20;                              // 64*16384 per batch
    int r  = idx & 1048575;
    int ch = r >> 14;
    int hw = r & 16383;
    size_t nh = ((size_t)(bn << 14) + hw) * 64 + ch;
    out[idx] = a[nh] + b[nh] + c[nh];
}

// ---------------------------------------------------------------------------
extern "C" void kernel_launch(void* const* d_in, const int* in_sizes, int n_in,
                              void* d_out, int out_size, void* d_ws, size_t ws_size,
                              hipStream_t stream) {
    const float* x        = (const float*)d_in[0];   // (4,64,128,128)
    const float* offset_w = (const float*)d_in[1];   // (7,18,64,3,3)
    const float* offset_b = (const float*)d_in[2];   // (7,18)
    const float* deform_w = (const float*)d_in[3];   // (7,64,64,3,3)
    const float* gamma    = (const float*)d_in[4];   // (7,64)
    const float* beta     = (const float*)d_in[5];   // (7,64)

    // workspace layout (needs ~72 MiB)
    char* ws = (char*)d_ws;
    float* bufA    = (float*)(ws);                          // 16 MiB NHWC
    float* bufB    = (float*)(ws + ((size_t)16 << 20));
    float* bufC    = (float*)(ws + ((size_t)32 << 20));
    float* bufD    = (float*)(ws + ((size_t)48 << 20));
    float* offbuf  = (float*)(ws + ((size_t)64 << 20));     // 4.5 MiB [pid][18]
    float* packedW = (float*)(ws + ((size_t)69 << 20));     // ~1 MiB, 7 layers
    float* stats   = (float*)(ws + ((size_t)71 << 20));     // 128 floats

    // one-time prep
    pack_weights_kernel<<<1008, 256, 0, stream>>>(deform_w, packedW);
    nchw_to_nhwc_kernel<<<16384, 256, 0, stream>>>(x, bufA);

    auto run_layer = [&](int li, const float* src, float* dst) {
        zero_stats_kernel<<<1, 128, 0, stream>>>(stats);
        offset_conv_kernel<<<256, 256, 0, stream>>>(
            src, offset_w + (size_t)li * 18 * 64 * 9, offset_b + (size_t)li * 18, offbuf);
        deform_wmma_kernel<<<512, 256, 0, stream>>>(
            src, offbuf, packedW + (size_t)li * 36864, dst, stats);
        norm_relu_kernel<<<16384, 256, 0, stream>>>(
            dst, stats, gamma + (size_t)li * 64, beta + (size_t)li * 64);
    };

    // DAG: out1 = L3(L2(s)); out2 = L5(s); out3 = L6(s); s = L1(L0(x)) + L4(x)
    run_layer(0, bufA, bufB);
    run_layer(1, bufB, bufC);
    run_layer(4, bufA, bufB);
    add_kernel<<<4096, 256, 0, stream>>>(bufC, bufB, bufA);   // s -> bufA
    run_layer(2, bufA, bufB);
    run_layer(3, bufB, bufC);
    run_layer(5, bufA, bufB);
    run_layer(6, bufA, bufD);
    combine_out_kernel<<<16384, 256, 0, stream>>>(bufC, bufB, bufD, (float*)d_out);
}